// PlenoxelModel_9543417332050
// MI455X (gfx1250) — compile-verified
//
#include <hip/hip_runtime.h>
#include <hip/hip_bf16.h>

typedef __attribute__((ext_vector_type(2))) float v2f;
typedef __attribute__((ext_vector_type(8))) float v8f;

#define GRID_N 128

// One thread per point. 256 threads = 8 wave32.
// Phase 1: gather voxel row (7x global_load_b128), compute SH basis + sigma;
//          stage SH + voxel row in LDS (with zero-pad slots).
// Phase 2: per wave, per 16-point group: color via chained V_WMMA_F32_16X16X4_F32,
//          taking the diagonal of D = A(sh 16x4) * B(K 4x16). No EXEC manipulation:
//          pad handling is done with clamped LDS indices into known-zero slots.
__launch_bounds__(256)
__global__ void plenoxel_fused_kernel(const float* __restrict__ xin,
                                      const float* __restrict__ din,
                                      const float* __restrict__ grid,
                                      float* __restrict__ out,
                                      int P)
{
    // Row strides 10 / 30 / 3 floats: 32 lanes map to 32 distinct banks (mod 64)
    // for both write (stride*lane) and read (stride*(lane&15)) phases.
    // shL[..][9] == 0 and kL[..][28..29] == 0 are the WMMA K-pad zero slots.
    __shared__ __align__(16) float shL[8][32][10];
    __shared__ __align__(16) float kL [8][32][30];
    __shared__ __align__(16) float colL[8][32][3];

    const int tid  = threadIdx.x;
    const int wave = tid >> 5;
    const int lane = tid & 31;
    const int pt   = blockIdx.x * 256 + tid;
    const bool valid = pt < P;
    const int ptc  = valid ? pt : (P - 1);   // clamped load index keeps EXEC full later

    // ---------------- Phase 1: gather + SH basis ----------------
    // x/d are read-once (48 MB): non-temporal, keep L2 for the voxel grid.
    const float px = __builtin_nontemporal_load(xin + 3 * ptc + 0);
    const float py = __builtin_nontemporal_load(xin + 3 * ptc + 1);
    const float pz = __builtin_nontemporal_load(xin + 3 * ptc + 2);
    const bool mask = (fabsf(px) < 1.f) && (fabsf(py) < 1.f) && (fabsf(pz) < 1.f);

    // idx = clip(int(x * N/2 + N/2), 0, N-1); clamp in float to keep the cast defined
    const int i0 = (int)fminf(fmaxf(px * 64.f + 64.f, 0.f), 127.f);
    const int i1 = (int)fminf(fmaxf(py * 64.f + 64.f, 0.f), 127.f);
    const int i2 = (int)fminf(fmaxf(pz * 64.f + 64.f, 0.f), 127.f);
    const int v  = (i0 * GRID_N + i1) * GRID_N + i2;

    // 28 floats, 112 B, 16 B aligned: 7x global_load_b128. Regular-temporal:
    // the 235 MB grid has ~2x reuse against the 192 MB L2 -> worth caching.
    const float4* __restrict__ vox = (const float4*)(grid + (long)v * 28);
    float4 t[7];
#pragma unroll
    for (int i = 0; i < 7; ++i) t[i] = vox[i];

    const float sigma = mask ? fmaxf(t[0].x, 0.f) : 0.f;
    if (valid) __builtin_nontemporal_store(sigma, out + (long)3 * P + pt);

    // Stage voxel row (incl. raw sigma slot) into LDS: 14x ds_store_b64
    // (row base = lane*120 B, 8 B aligned), then zero the two pad slots.
    {
        float2* krow2 = (float2*)&kL[wave][lane][0];
        const float2* t2 = (const float2*)&t[0];
#pragma unroll
        for (int i = 0; i < 14; ++i) krow2[i] = t2[i];
        krow2[14] = make_float2(0.f, 0.f);   // kL[..][28..29] = 0 (WMMA pad)
    }

    // SH basis (l<=2, 9 terms) + zero pad slot
    const float dx = __builtin_nontemporal_load(din + 3 * ptc + 0);
    const float dy = __builtin_nontemporal_load(din + 3 * ptc + 1);
    const float dz = __builtin_nontemporal_load(din + 3 * ptc + 2);
    float sb[10];
    sb[0] =  0.28209479177387814f;
    sb[1] = -0.4886025119029199f * dy;
    sb[2] =  0.4886025119029199f * dz;
    sb[3] = -0.4886025119029199f * dx;
    sb[4] =  1.0925484305920792f * dx * dy;
    sb[5] = -1.0925484305920792f * dy * dz;
    sb[6] =  0.31539156525252005f * (2.f * dz * dz - dx * dx - dy * dy);
    sb[7] = -1.0925484305920792f * dx * dz;
    sb[8] =  0.5462742152960396f * (dx * dx - dy * dy);
    sb[9] =  0.f;                            // shL[..][9] = 0 (WMMA pad)
#pragma unroll
    for (int j = 0; j < 10; ++j) shL[wave][lane][j] = sb[j];

    __syncthreads();

    // ---------------- Phase 2: WMMA batched 3x9 matvec ----------------
    // A (16x4 f32): lane m(0..15) row m; v0 holds K=j0, v1 holds K=j0+1,
    //               lanes 16..31 hold K=j0+2 / j0+3 (ISA 32-bit A layout).
    // B (4x16 f32): N = lane&15; lane-half selects K half, VGPR selects K within half.
    // Diagonal of D(16x16) = the 16 per-point dot products.
    const int m  = lane & 15;
    const int kh = lane >> 4;   // lane half

    // Loop-invariant diagonal-extraction predicates:
    // (p,p): p<8 -> acc[p] @ lane p ; p>=8 -> acc[p-8] @ lane p+16
    const int r = (lane < 8) ? lane : (lane - 24);
    const bool e1 = (r == 1), e2 = (r == 2), e3 = (r == 3), e4 = (r == 4),
               e5 = (r == 5), e6 = (r == 6), e7 = (r == 7);
    const bool dlo = (lane < 8), dhi = (lane >= 24);

#pragma unroll
    for (int g = 0; g < 2; ++g) {
        const float* __restrict__ shRow = &shL[wave][g * 16 + m][0];
        const float* __restrict__ kRow  = &kL [wave][g * 16 + m][0];
#pragma unroll
        for (int ch = 0; ch < 3; ++ch) {
            v8f acc = {};
#pragma unroll
            for (int c = 0; c < 3; ++c) {
                const int j0 = 4 * c + 2 * kh;
                const int j1 = j0 + 1;
                // Clamped indices into known-zero pad slots: loads stay
                // unconditional -> no EXEC manipulation between WMMAs.
                const int ia0 = (j0 < 9) ? j0 : 9;
                const int ia1 = (j1 < 9) ? j1 : 9;
                const int ib0 = (j0 < 9) ? (1 + 9 * ch + j0) : 28;
                const int ib1 = (j1 < 9) ? (1 + 9 * ch + j1) : 28;
                v2f a; a.x = shRow[ia0]; a.y = shRow[ia1];
                v2f b; b.x = kRow[ib0];  b.y = kRow[ib1];
                // D = A*B + C, exact f32 accumulate across the 3 K-chunks
                acc = __builtin_amdgcn_wmma_f32_16x16x4_f32(
                        /*neg_a=*/false, a, /*neg_b=*/false, b,
                        /*c_mod=*/(short)0, acc,
                        /*reuse_a=*/false, /*reuse_b=*/false);
            }
            float dv = acc[0];
            dv = e1 ? acc[1] : dv;
            dv = e2 ? acc[2] : dv;
            dv = e3 ? acc[3] : dv;
            dv = e4 ? acc[4] : dv;
            dv = e5 ? acc[5] : dv;
            dv = e6 ? acc[6] : dv;
            dv = e7 ? acc[7] : dv;
            if (dlo)      colL[wave][g * 16 + lane][ch] = dv;
            else if (dhi) colL[wave][g * 16 + (lane - 16)][ch] = dv;
        }
    }

    __syncthreads();

    // ---------------- Phase 3: masked color store (write-once -> NT) ----------------
    float c0 = colL[wave][lane][0];
    float c1 = colL[wave][lane][1];
    float c2 = colL[wave][lane][2];
    if (!mask) { c0 = 0.f; c1 = 0.f; c2 = 0.f; }
    if (valid) {
        __builtin_nontemporal_store(c0, out + 3 * pt + 0);
        __builtin_nontemporal_store(c1, out + 3 * pt + 1);
        __builtin_nontemporal_store(c2, out + 3 * pt + 2);
    }
}

extern "C" void kernel_launch(void* const* d_in, const int* in_sizes, int n_in,
                              void* d_out, int out_size, void* d_ws, size_t ws_size,
                              hipStream_t stream) {
    const float* x    = (const float*)d_in[0];
    const float* d    = (const float*)d_in[1];
    const float* grid = (const float*)d_in[2];
    float* out        = (float*)d_out;
    const int P = in_sizes[0] / 3;          // 4194304
    const int blocks = (P + 255) / 256;     // 16384
    plenoxel_fused_kernel<<<blocks, 256, 0, stream>>>(x, d, grid, out, P);
}